// CustomGConvLSTM_31722628448355
// MI455X (gfx1250) — compile-verified
//
#include <hip/hip_runtime.h>
#include <math.h>

// GConvLSTM with zero initial state:
//   cheb(H=0) == bh  -> Wh0/Wh1 unused;  C=0 -> forget gate + wc[0],wc[1] unused.
// Per period p: G_g = Xp@Wx0[g] + (Lhat Xp)@Wx1[g] + (bx+bh+bg)[g],  g in {i=0,c=2,o=3}
//   I=sig(G_i); T=tanh(G_c); Cn=I*T; O=sig(G_o + wc[2]*Cn); out += O*tanh(Cn)

typedef float v2f __attribute__((ext_vector_type(2)));
typedef float v8f __attribute__((ext_vector_type(8)));

#define F 64
#define PERIODS 8
#define NGATES 3            // i, c, o (forget gate dead: C=0)
#define KSTEPS 16           // 64 / K(=4)
#define NTILES 4            // 64 / 16 output columns
#define WSLOT (KSTEPS * NTILES * 32)   // float4 entries per gate = 2048 (32 KB)

__device__ __forceinline__ float sigmoidf_(float x) {
  return 1.0f / (1.0f + __expf(-x));
}

// ---- 1. out-degree: deg[src] += w ----
__global__ void deg_kernel(const int* __restrict__ ei, const float* __restrict__ ew,
                           float* __restrict__ deg, int E) {
  int e = blockIdx.x * blockDim.x + threadIdx.x;
  if (e < E) atomicAdd(&deg[ei[e]], ew[e]);
}

// ---- 2. dis = deg>0 ? rsqrt(max(deg,1e-12)) : 0  (in place) ----
__global__ void dis_kernel(float* deg, int N) {
  int n = blockIdx.x * blockDim.x + threadIdx.x;
  if (n < N) {
    float d = deg[n];
    deg[n] = (d > 0.f) ? rsqrtf(fmaxf(d, 1e-12f)) : 0.f;
  }
}

// ---- 3. SpMM: TX[p][dst][f] += wn_e * X[src][f][p]  (thread = (edge, feature)) ----
__global__ void spmm_kernel(const float* __restrict__ X, const int* __restrict__ ei,
                            const float* __restrict__ ew, const float* __restrict__ dis,
                            float* __restrict__ TX, int E, int N) {
  int tid = blockIdx.x * blockDim.x + threadIdx.x;
  int e = tid >> 6;
  int f = tid & 63;
  if (e >= E) return;
  int src = ei[e];
  int dst = ei[E + e];
  float w = -dis[src] * ew[e] * dis[dst];
  // 8 periods of X[src, f, :] are contiguous (32 bytes)
  const float4* xp = (const float4*)(X + (size_t)src * (F * PERIODS) + f * PERIODS);
  float4 a = xp[0];
  float4 b = xp[1];
  float vals[8] = { a.x, a.y, a.z, a.w, b.x, b.y, b.z, b.w };
  size_t base  = (size_t)dst * F + f;
  size_t plane = (size_t)N * F;
#pragma unroll
  for (int p = 0; p < PERIODS; ++p)
    atomicAdd(&TX[(size_t)p * plane + base], w * vals[p]);
}

// ---- 3b. pack weights into WMMA B-fragment lane order ----
// Wp[slot][ks][nt][lane] = float4{ Wx0[g][k+0][col], Wx0[g][k+1][col],
//                                  Wx1[g][k+0][col], Wx1[g][k+1][col] }
// with mrow=lane&15, khalf=(lane>>4)*2, k=ks*4+khalf, col=nt*16+mrow, slot->g in {0,2,3}
__global__ void pack_weights(const float* __restrict__ Wx0, const float* __restrict__ Wx1,
                             float* __restrict__ Wp) {
  int tid = blockIdx.x * blockDim.x + threadIdx.x;
  if (tid >= NGATES * WSLOT) return;
  int lane = tid & 31;
  int nt   = (tid >> 5) & 3;
  int ks   = (tid >> 7) & 15;
  int slot = tid >> 11;
  const int gmap[NGATES] = {0, 2, 3};
  int g    = gmap[slot];
  int mrow = lane & 15;
  int k    = ks * 4 + ((lane >> 4) << 1);
  int col  = nt * 16 + mrow;
  const float* W0 = Wx0 + (size_t)g * F * F;
  const float* W1 = Wx1 + (size_t)g * F * F;
  float4 v;
  v.x = W0[(k + 0) * F + col];
  v.y = W0[(k + 1) * F + col];
  v.z = W1[(k + 0) * F + col];
  v.w = W1[(k + 1) * F + col];
  ((float4*)Wp)[tid] = v;
}

// ---- 4. fused FP32-WMMA GEMMs + LSTM gates + period-sum ----
// Block = 16-node tile, 8 waves, wave w handles period w.
// LDS: 32 KB X tile + 32 KB TX tiles + 32 KB packed gate weights = 96 KB (WGP has 320 KB).
__global__ void __launch_bounds__(256)
fused_wmma_lstm(const float* __restrict__ X, const float* __restrict__ TX,
                const float* __restrict__ Wp,
                const float* __restrict__ bx, const float* __restrict__ bh,
                const float* __restrict__ wc, const float* __restrict__ bg,
                float* __restrict__ out, int N) {
  __shared__ float ldsX[16 * F * PERIODS];   // 8192 floats
  __shared__ float ldsT[PERIODS * 16 * F];   // 8192 floats
  __shared__ float ldsW[WSLOT * 4];          // 8192 floats (one gate's packed B frags)

  const int tid  = threadIdx.x;
  const int wave = tid >> 5;                 // period
  const int lane = tid & 31;
  const int n0   = blockIdx.x * 16;
  const int nrows = (N - n0 < 16) ? (N - n0) : 16;

  // stage X tile (contiguous in memory)
  {
    const float4* g = (const float4*)(X + (size_t)n0 * (F * PERIODS));
    float4* s = (float4*)ldsX;
    const int tot = nrows * (F * PERIODS) / 4;
    for (int i = tid; i < tot; i += 256) s[i] = g[i];
  }
  // stage this wave's TX tile
  {
    const float4* g = (const float4*)(TX + ((size_t)wave * N + n0) * F);
    float4* s = (float4*)(ldsT + wave * 16 * F);
    const int tot = nrows * F / 4;
    for (int i = lane; i < tot; i += 32) s[i] = g[i];
  }
  // stage gate-0 packed weights
  {
    const float4* g = (const float4*)Wp;
    float4* s = (float4*)ldsW;
    for (int i = tid; i < WSLOT; i += 256) s[i] = g[i];
  }
  __syncthreads();

  // ISA 16x4 f32 A layout: lanes 0-15 -> M=lane, K = ks*4 + {0,1};
  //                        lanes 16-31 -> M=lane-16, K = ks*4 + {2,3}.
  const int mrow  = lane & 15;
  const int khalf = (lane >> 4) << 1;        // 0 or 2

  v2f ax[KSTEPS], at[KSTEPS];
#pragma unroll
  for (int ks = 0; ks < KSTEPS; ++ks) {
    int k = ks * 4 + khalf;
    ax[ks].x = ldsX[mrow * (F * PERIODS) + (k + 0) * PERIODS + wave];
    ax[ks].y = ldsX[mrow * (F * PERIODS) + (k + 1) * PERIODS + wave];
    at[ks].x = ldsT[wave * 16 * F + mrow * F + (k + 0)];
    at[ks].y = ldsT[wave * 16 * F + mrow * F + (k + 1)];
  }

  const v8f vzero = {0.f, 0.f, 0.f, 0.f, 0.f, 0.f, 0.f, 0.f};
  v8f acc[NGATES][NTILES];

  for (int s = 0; s < NGATES; ++s) {
#pragma unroll
    for (int nt = 0; nt < NTILES; ++nt) acc[s][nt] = vzero;
#pragma unroll
    for (int ks = 0; ks < KSTEPS; ++ks) {
#pragma unroll
      for (int nt = 0; nt < NTILES; ++nt) {
        // one ds_load_b128: {Wx0[k][col], Wx0[k+1][col], Wx1[k][col], Wx1[k+1][col]}
        float4 w = ((const float4*)ldsW)[ks * (NTILES * 32) + nt * 32 + lane];
        v2f b0 = {w.x, w.y};
        v2f b1 = {w.z, w.w};
        acc[s][nt] = __builtin_amdgcn_wmma_f32_16x16x4_f32(false, ax[ks], false, b0,
                                                           (short)0, acc[s][nt], false, false);
        acc[s][nt] = __builtin_amdgcn_wmma_f32_16x16x4_f32(false, at[ks], false, b1,
                                                           (short)0, acc[s][nt], false, false);
      }
    }
    if (s < NGATES - 1) {
      __syncthreads();   // all waves finished reading current gate's weights
      const float4* g = (const float4*)(Wp + (size_t)(s + 1) * WSLOT * 4);
      float4* sm = (float4*)ldsW;
      for (int i = tid; i < WSLOT; i += 256) sm[i] = g[i];
      __syncthreads();
    }
  }

  // C/D layout: VGPR v, lanes 0-15 -> M=v, lanes 16-31 -> M=v+8; col = lane%16
#pragma unroll
  for (int nt = 0; nt < NTILES; ++nt) {
    int col = nt * 16 + mrow;
    float Bi = bx[0 * F + col] + bh[0 * F + col] + bg[0 * F + col];
    float Bc = bx[2 * F + col] + bh[2 * F + col] + bg[2 * F + col];
    float Bo = bx[3 * F + col] + bh[3 * F + col] + bg[3 * F + col];
    float w2 = wc[2 * F + col];
#pragma unroll
    for (int v = 0; v < 8; ++v) {
      float I   = sigmoidf_(acc[0][nt][v] + Bi);
      float T   = tanhf(acc[1][nt][v] + Bc);
      float Cn  = I * T;
      float O   = sigmoidf_(acc[2][nt][v] + Bo + w2 * Cn);
      float val = O * tanhf(Cn);
      int node  = n0 + v + ((lane >> 4) << 3);
      if (node < N)
        atomicAdd(&out[(size_t)node * F + col], val);  // sum over periods
    }
  }
}

extern "C" void kernel_launch(void* const* d_in, const int* in_sizes, int n_in,
                              void* d_out, int out_size, void* d_ws, size_t ws_size,
                              hipStream_t stream) {
  const float* X   = (const float*)d_in[0];
  const int*   ei  = (const int*)d_in[1];
  const float* ew  = (const float*)d_in[2];
  const float* Wx0 = (const float*)d_in[3];
  const float* Wx1 = (const float*)d_in[4];
  const float* bx  = (const float*)d_in[5];
  // d_in[6]=Wh0, d_in[7]=Wh1: unused (H=0 makes cheb(H)=bh exactly)
  const float* bh  = (const float*)d_in[8];
  const float* wc  = (const float*)d_in[9];
  const float* bg  = (const float*)d_in[10];
  float* out = (float*)d_out;

  const int N = in_sizes[0] / (F * PERIODS);
  const int E = in_sizes[2];

  float* deg = (float*)d_ws;                       // N floats
  float* TX  = deg + N;                            // PERIODS*N*F floats
  float* Wp  = TX + (size_t)PERIODS * N * F;       // NGATES*WSLOT*4 floats (96 KB)

  size_t zero_floats = (size_t)N + (size_t)PERIODS * N * F;
  hipMemsetAsync(d_ws, 0, zero_floats * sizeof(float), stream);
  hipMemsetAsync(d_out, 0, (size_t)N * F * sizeof(float), stream);

  deg_kernel<<<(E + 255) / 256, 256, 0, stream>>>(ei, ew, deg, E);
  dis_kernel<<<(N + 255) / 256, 256, 0, stream>>>(deg, N);
  pack_weights<<<(NGATES * WSLOT + 255) / 256, 256, 0, stream>>>(Wx0, Wx1, Wp);

  long long spmm_threads = (long long)E * 64;
  spmm_kernel<<<(unsigned)((spmm_threads + 255) / 256), 256, 0, stream>>>(
      X, ei, ew, deg, TX, E, N);

  fused_wmma_lstm<<<(N + 15) / 16, 256, 0, stream>>>(
      X, TX, Wp, bx, bh, wc, bg, out, N);
}